// GraniteMoeTopKGating_43963285242549
// MI455X (gfx1250) — compile-verified
//
#include <hip/hip_runtime.h>
#include <hip/hip_bf16.h>
#include <math.h>

typedef __attribute__((ext_vector_type(2))) float v2f;
typedef __attribute__((ext_vector_type(8))) float v8f;

#define TOKENS       16384
#define HIDDEN       4096
#define NUM_EXPERTS  64
#define TOP_K        8
#define WAVES_PER_BLOCK 4
#define TOKENS_PER_WAVE 16
#define LDS_ROW      (NUM_EXPERTS + 1)   // pad to avoid 16-way bank conflicts on column reads

__global__ __launch_bounds__(WAVES_PER_BLOCK * 32)
void moe_router_topk_kernel(const float* __restrict__ hidden,
                            const float* __restrict__ weight,
                            float* __restrict__ out_idx,
                            float* __restrict__ out_gates)
{
    __shared__ float lds_logits[WAVES_PER_BLOCK][TOKENS_PER_WAVE][LDS_ROW];

    const int wave = threadIdx.x >> 5;
    const int lane = threadIdx.x & 31;
    const int tile = blockIdx.x * WAVES_PER_BLOCK + wave;   // 16-token tile id
    const int t0   = tile * TOKENS_PER_WAVE;

    // WMMA f32 16x16x4 fragment addressing (wave32):
    //  A  VGPR0/1: lanes 0-15 -> (M=lane, K=k+0 / k+1); lanes 16-31 -> (M=lane-16, K=k+2 / k+3)
    //  B  VGPR0/1: lanes 0-15 -> (K=k+0 / k+1, N=lane); lanes 16-31 -> (K=k+2 / k+3, N=lane-16)
    // Both fragments are a per-lane float2 at column offset koff = 2*(lane>>4).
    const int row  = lane & 15;
    const int koff = (lane >> 4) << 1;

    const float* __restrict__ aptr = hidden + (size_t)(t0 + row) * HIDDEN + koff;
    const float* __restrict__ b0   = weight + (size_t)( 0 + row) * HIDDEN + koff;
    const float* __restrict__ b1   = weight + (size_t)(16 + row) * HIDDEN + koff;
    const float* __restrict__ b2   = weight + (size_t)(32 + row) * HIDDEN + koff;
    const float* __restrict__ b3   = weight + (size_t)(48 + row) * HIDDEN + koff;

    v8f acc0 = {}, acc1 = {}, acc2 = {}, acc3 = {};

#pragma unroll 4
    for (int k = 0; k < HIDDEN; k += 4) {
        v2f a  = *(const v2f*)(aptr + k);
        v2f w0 = *(const v2f*)(b0 + k);
        v2f w1 = *(const v2f*)(b1 + k);
        v2f w2 = *(const v2f*)(b2 + k);
        v2f w3 = *(const v2f*)(b3 + k);
        // 8-arg form: (neg_a, A, neg_b, B, c_mod, C, reuse_a, reuse_b)
        acc0 = __builtin_amdgcn_wmma_f32_16x16x4_f32(false, a, false, w0, (short)0, acc0, false, false);
        acc1 = __builtin_amdgcn_wmma_f32_16x16x4_f32(false, a, false, w1, (short)0, acc1, false, false);
        acc2 = __builtin_amdgcn_wmma_f32_16x16x4_f32(false, a, false, w2, (short)0, acc2, false, false);
        acc3 = __builtin_amdgcn_wmma_f32_16x16x4_f32(false, a, false, w3, (short)0, acc3, false, false);
    }

    // C/D layout: VGPR i, lanes 0-15 -> (M=i, N=lane); lanes 16-31 -> (M=8+i, N=lane-16)
    const int mbase = (lane >> 4) << 3;      // 0 or 8
    const int ecol  = lane & 15;
#pragma unroll
    for (int i = 0; i < 8; ++i) {
        lds_logits[wave][mbase + i][ 0 + ecol] = acc0[i];
        lds_logits[wave][mbase + i][16 + ecol] = acc1[i];
        lds_logits[wave][mbase + i][32 + ecol] = acc2[i];
        lds_logits[wave][mbase + i][48 + ecol] = acc3[i];
    }

    __syncthreads();

    // Lanes 0-15: per-token top-8 (iterative masked argmax; strict '>' keeps the
    // lowest index on ties, matching jax.lax.top_k), then softmax over the 8 kept.
    if (lane < 16) {
        const int token = t0 + lane;
        const float* lg = lds_logits[wave][lane];

        float v[NUM_EXPERTS];
#pragma unroll
        for (int j = 0; j < NUM_EXPERTS; ++j) v[j] = lg[j];

        unsigned long long taken = 0ull;
        float kept[TOP_K];
        int   kidx[TOP_K];
#pragma unroll
        for (int s = 0; s < TOP_K; ++s) {
            float best = -INFINITY;
            int   bi   = 0;
#pragma unroll
            for (int j = 0; j < NUM_EXPERTS; ++j) {
                bool used = (taken >> j) & 1ull;
                if (!used && v[j] > best) { best = v[j]; bi = j; }
            }
            taken |= 1ull << bi;
            kept[s] = best;
            kidx[s] = bi;
        }

        // softmax over the kept logits; kept[0] is the max (descending order)
        float mx = kept[0];
        float sum = 0.f;
        float e[TOP_K];
#pragma unroll
        for (int s = 0; s < TOP_K; ++s) { e[s] = __expf(kept[s] - mx); sum += e[s]; }
        float inv = 1.f / sum;

        float* __restrict__ oi = out_idx   + (size_t)token * TOP_K;
        float* __restrict__ og = out_gates + (size_t)token * TOP_K;
#pragma unroll
        for (int s = 0; s < TOP_K; ++s) {
            oi[s] = (float)kidx[s];
            og[s] = e[s] * inv;
        }
    }
}

extern "C" void kernel_launch(void* const* d_in, const int* in_sizes, int n_in,
                              void* d_out, int out_size, void* d_ws, size_t ws_size,
                              hipStream_t stream) {
    const float* hidden = (const float*)d_in[0];   // [TOKENS, HIDDEN] f32
    const float* weight = (const float*)d_in[1];   // [NUM_EXPERTS, HIDDEN] f32
    float* out       = (float*)d_out;
    float* out_idx   = out;                                  // [TOKENS, TOP_K] (indices as floats)
    float* out_gates = out + (size_t)TOKENS * TOP_K;         // [TOKENS, TOP_K] gates

    const int blocks = TOKENS / (WAVES_PER_BLOCK * TOKENS_PER_WAVE); // 256
    moe_router_topk_kernel<<<blocks, WAVES_PER_BLOCK * 32, 0, stream>>>(
        hidden, weight, out_idx, out_gates);
}